// DetectorWithNMS_27771258536186
// MI455X (gfx1250) — compile-verified
//
#include <hip/hip_runtime.h>
#include <stdint.h>

#define N_BOX   8400
#define N_PAD   16384          // next pow2 for bitonic sort
#define C_ALL   85
#define N_CLS   80
#define CONF_THR 0.5f
#define NMS_THR  0.3f
#define DEC_ROWS 256           // rows per decode tile / block

typedef unsigned int u32x4 __attribute__((ext_vector_type(4)));
typedef unsigned int u32x8 __attribute__((ext_vector_type(8)));

// Low 32 bits of a generic pointer to LDS == raw LDS byte offset (flat-address
// LDS aperture layout, ISA 10.2).
__device__ __forceinline__ unsigned lds_low32(const void* p) {
    return (unsigned)(unsigned long long)p;
}

// ---------------- workspace layout (bytes) ----------------
#define WS_KEYS_OFF  0                                // 16384 * 8 = 131072
#define WS_BOX_OFF   (N_PAD * 8)                      // 8400 * 16 = 134400
#define WS_CONF_OFF  (WS_BOX_OFF + N_BOX * 16)        // 8400 * 4
#define WS_CAT_OFF   (WS_CONF_OFF + N_BOX * 4)        // 8400 * 4

// =====================================================================
// Kernel 1: decode. One block = one 256-row tile, staged into LDS by the
// Tensor Data Mover, then one thread per box computes box/conf/argmax and
// the 64-bit stable sort key.
// =====================================================================
__global__ void decode_kernel(const float* __restrict__ x,
                              unsigned long long* __restrict__ keys,
                              float4* __restrict__ boxes,
                              float* __restrict__ conf_out,
                              int* __restrict__ cat_out) {
    extern __shared__ float lds[];     // DEC_ROWS * 85 floats = 87040 B
    const int tid  = threadIdx.x;
    const int row0 = blockIdx.x * DEC_ROWS;
    int rows = N_BOX - row0;
    if (rows > DEC_ROWS) rows = DEC_ROWS;

    if (rows > 0) {
        if (tid < 32) {  // one wave issues the TDM DMA (EXEC ignored by TDM)
            unsigned long long ga =
                (unsigned long long)(const void*)(x + (size_t)row0 * C_ALL);
            unsigned la = lds_low32(lds);
            // ---- D# group 0 (128b): count=1, lds_addr, global_addr, type=2
            u32x4 g0;
            g0[0] = 1u;
            g0[1] = la;
            g0[2] = (unsigned)(ga & 0xFFFFFFFFu);
            g0[3] = ((unsigned)(ga >> 32) & 0x01FFFFFFu) | 0x80000000u;
            // ---- D# group 1 (256b): 2D tensor, 4-byte elements
            u32x8 g1;
            g1[0] = 2u << 16;                     // data_size = 4B
            g1[1] = ((unsigned)C_ALL) << 16;      // tensor_dim0[15:0] = 85
            g1[2] = ((unsigned)rows)  << 16;      // dim0 hi=0 | tensor_dim1 lo = rows
            g1[3] = ((unsigned)C_ALL) << 16;      // dim1 hi=0 | tile_dim0 = 85
            g1[4] = (unsigned)rows;               // tile_dim1 = rows, tile_dim2 = 0
            g1[5] = (unsigned)C_ALL;              // tensor_dim0_stride = 85
            g1[6] = 0u;
            g1[7] = 0u;
            asm volatile("tensor_load_to_lds %0, %1"
                         :: "s"(g0), "s"(g1) : "memory");
            __builtin_amdgcn_s_wait_tensorcnt(0);
        }
    }
    __syncthreads();   // publish the LDS tile to all 8 waves

    const int r = row0 + tid;
    if (r < N_BOX) {
        const float* p = &lds[tid * C_ALL];   // stride 85: conflict-free
        float v0 = p[0], v1 = p[1], v2 = p[2], v3 = p[3], obj = p[4];

        int s, dim, li;
        if (r < 6400)      { s = 8;  dim = 80; li = r;        }
        else if (r < 8000) { s = 16; dim = 40; li = r - 6400; }
        else               { s = 32; dim = 20; li = r - 8000; }
        float gx = (float)(li % dim);
        float gy = (float)(li / dim);
        float fs = (float)s;

        float cx = (v0 + gx) * fs;
        float cy = (v1 + gy) * fs;
        float w  = expf(v2) * fs;
        float h  = expf(v3) * fs;

        float maxv = p[5];
        int   arg  = 0;
        #pragma unroll 5
        for (int c = 1; c < N_CLS; ++c) {
            float v = p[5 + c];
            if (v > maxv) { maxv = v; arg = c; }   // first-max on ties
        }
        float conf = obj * maxv;

        float4 bb;
        bb.x = cx - 0.5f * w;  bb.y = cy - 0.5f * h;
        bb.z = cx + 0.5f * w;  bb.w = cy + 0.5f * h;
        boxes[r]    = bb;
        conf_out[r] = conf;
        cat_out[r]  = arg;

        // Stable sort key: hi32 = monotone map of (-conf | +inf), lo32 = index
        float f = (conf > CONF_THR) ? -conf : __uint_as_float(0x7F800000u);
        unsigned u = __float_as_uint(f);
        unsigned m = (u & 0x80000000u) ? ~u : (u | 0x80000000u);
        keys[r] = ((unsigned long long)m << 32) | (unsigned)r;
    } else {
        keys[r] = ~0ull;   // padding: sorts last (launch covers exactly N_PAD)
    }
}

// =====================================================================
// Kernel 2: single-workgroup bitonic sort of 16384 u64 keys in LDS (128 KB).
// =====================================================================
__global__ void sort_kernel(unsigned long long* __restrict__ keys) {
    extern __shared__ unsigned long long sk[];
    const int nt = blockDim.x;
    for (int i = threadIdx.x; i < N_PAD; i += nt) sk[i] = keys[i];
    __syncthreads();
    for (int k = 2; k <= N_PAD; k <<= 1) {
        for (int j = k >> 1; j > 0; j >>= 1) {
            for (int i = threadIdx.x; i < N_PAD; i += nt) {
                int l = i ^ j;
                if (l > i) {
                    unsigned long long a = sk[i], b = sk[l];
                    bool up = ((i & k) == 0);
                    if ((a > b) == up) { sk[i] = b; sk[l] = a; }
                }
            }
            __syncthreads();
        }
    }
    for (int i = threadIdx.x; i < N_PAD; i += nt) keys[i] = sk[i];
}

// =====================================================================
// Kernel 3: greedy NMS + masked output, all state LDS-resident (~205 KB).
// =====================================================================
__global__ void nms_kernel(const unsigned long long* __restrict__ keys,
                           const float4* __restrict__ boxes,
                           const float* __restrict__ conf,
                           const int* __restrict__ cats,
                           float* __restrict__ out) {
    extern __shared__ unsigned char smem[];
    float4*        b4   = (float4*)smem;                       // [0, 134400)
    float*         cf   = (float*)(smem + N_BOX * 16);         // 33600
    int*           ct   = (int*)  (smem + N_BOX * 20);         // 33600
    unsigned char* supp = smem + N_BOX * 24;                   // 8400
    int*           nvp  = (int*)  (smem + N_BOX * 24 + 8416);

    const int tid = threadIdx.x;
    const int nt  = blockDim.x;

    if (tid == 0) *nvp = 0;
    __syncthreads();

    // Gather into sorted-rank order.
    int local = 0;
    for (int r = tid; r < N_BOX; r += nt) {
        unsigned long long k = keys[r];
        int i = (int)(k & 0xFFFFFFFFull);
        float  c = conf[i];
        b4[r]   = boxes[i];
        cf[r]   = c;
        ct[r]   = cats[i];
        supp[r] = 0;
        if (c > CONF_THR) ++local;
    }
    atomicAdd(nvp, local);
    __syncthreads();
    const int nv = *nvp;   // valid boxes occupy ranks [0, nv) after the sort

    // Sequential greedy scan; barrier only after iterations that wrote.
    bool wrote = false;
    for (int r = 0; r < nv; ++r) {
        if (wrote) { __syncthreads(); wrote = false; }
        if (supp[r]) continue;                    // uniform across block
        float4 br = b4[r];
        int    cr = ct[r];
        float  ar = (br.z - br.x) * (br.w - br.y);
        for (int j = r + 1 + tid; j < N_BOX; j += nt) {
            if (ct[j] != cr) continue;
            float4 bj = b4[j];
            float lx = fmaxf(br.x, bj.x);
            float ly = fmaxf(br.y, bj.y);
            float rx = fminf(br.z, bj.z);
            float ry = fminf(br.w, bj.w);
            float iw = fmaxf(rx - lx, 0.0f);
            float ih = fmaxf(ry - ly, 0.0f);
            float inter = iw * ih;
            float aj = (bj.z - bj.x) * (bj.w - bj.y);
            float iou = inter / (ar + aj - inter);
            if (iou > NMS_THR) supp[j] = 1;
        }
        wrote = true;
    }
    __syncthreads();

    // result[r] = [x1,y1,x2,y2,conf,cat] * keep
    for (int r = tid; r < N_BOX; r += nt) {
        float m = ((r < nv) && (supp[r] == 0)) ? 1.0f : 0.0f;
        float4 bb = b4[r];
        out[r * 6 + 0] = bb.x * m;
        out[r * 6 + 1] = bb.y * m;
        out[r * 6 + 2] = bb.z * m;
        out[r * 6 + 3] = bb.w * m;
        out[r * 6 + 4] = cf[r] * m;
        out[r * 6 + 5] = (float)ct[r] * m;
    }
}

// =====================================================================
extern "C" void kernel_launch(void* const* d_in, const int* in_sizes, int n_in,
                              void* d_out, int out_size, void* d_ws, size_t ws_size,
                              hipStream_t stream) {
    (void)in_sizes; (void)n_in; (void)out_size; (void)ws_size;
    const float* x = (const float*)d_in[0];
    unsigned char* ws = (unsigned char*)d_ws;
    unsigned long long* keys = (unsigned long long*)(ws + WS_KEYS_OFF);
    float4* boxes = (float4*)(ws + WS_BOX_OFF);
    float*  conf  = (float*) (ws + WS_CONF_OFF);
    int*    cats  = (int*)   (ws + WS_CAT_OFF);
    float*  out   = (float*)d_out;

    // 64 blocks * 256 threads = 16384 = N_PAD (covers key padding writes)
    decode_kernel<<<64, DEC_ROWS, DEC_ROWS * C_ALL * 4, stream>>>(
        x, keys, boxes, conf, cats);
    sort_kernel<<<1, 1024, N_PAD * 8, stream>>>(keys);
    nms_kernel<<<1, 1024, N_BOX * 24 + 8416 + 32, stream>>>(
        keys, boxes, conf, cats, out);
}